// GraphMemory_17660905521908
// MI455X (gfx1250) — compile-verified
//
#include <hip/hip_runtime.h>
#include <hip/hip_bf16.h>
#include <cstdint>
#include <cstddef>

#define E_DIM 384
#define W_DIM (E_DIM * E_DIM)            /* 147456 columns           */
#define KSLICE 96
#define NSLICE (E_DIM / KSLICE)          /* 4 k-slices               */
#define CHUNK_COLS 49152                 /* 151 MB bias+space -> L2  */
#define NCHUNK (W_DIM / CHUNK_COLS)      /* 3                        */
#define P2_TILE 4096                     /* cols per pass-2 block    */
#define INV_SQRT2 0.70710678118654752440f

#if defined(__gfx1250__) && __has_builtin(__builtin_amdgcn_global_load_async_to_lds_b128)
#define HAVE_ASYNC_LDS 1
#else
#define HAVE_ASYNC_LDS 0
#endif

#if HAVE_ASYNC_LDS
typedef int v4i __attribute__((vector_size(16)));
typedef __attribute__((address_space(1))) v4i* gv4i_p;   // global (AS1) int4*
typedef __attribute__((address_space(3))) v4i* lv4i_p;   // LDS    (AS3) int4*
#endif

// Order-preserving float -> uint mapping (for atomicMax on floats of any sign).
__device__ __forceinline__ unsigned enc_f32(float f) {
    unsigned u = __float_as_uint(f);
    return (u & 0x80000000u) ? ~u : (u | 0x80000000u);
}
__device__ __forceinline__ float dec_f32(unsigned k) {
    unsigned u = (k & 0x80000000u) ? (k & 0x7fffffffu) : ~k;
    return __uint_as_float(u);
}

__global__ void k_init(unsigned* __restrict__ keys) {
    int t = threadIdx.x;
    if (t < E_DIM) keys[t] = 0u;   // identity element for the ordered-uint max
}

// Pass 1: per-column online softmax stats over a 96-row k-slice.
__global__ __launch_bounds__(256) void k_pass1(
    const float* __restrict__ bias, const float* __restrict__ space,
    const float* __restrict__ kern, const float* __restrict__ x,
    float* __restrict__ pm, float* __restrict__ ps, int colBase)
{
    __shared__ float xs[KSLICE];
    const int tid = threadIdx.x;
    const int ks  = blockIdx.y * KSLICE;
    if (tid < KSLICE) xs[tid] = x[ks + tid];
    __syncthreads();

    const int col = colBase + blockIdx.x * 256 + tid;
    const int l   = col % E_DIM;           // diagonal row for this column
    const int j   = col / E_DIM;
    float extra = 0.0f;
    if (l >= ks && l < ks + KSLICE)
        extra = kern[(size_t)l * W_DIM + col] * x[j] * INV_SQRT2;

    float m = -3.0e38f, s = 0.0f;
    size_t rowIdx = (size_t)ks * W_DIM + col;
    for (int k0 = 0; k0 < KSLICE; k0 += 8) {
        float b[8], sp[8];
#pragma unroll
        for (int u = 0; u < 8; ++u) {
            size_t idx = rowIdx + (size_t)u * W_DIM;
            b[u]  = bias[idx];
            sp[u] = space[idx];
        }
        // speculative prefetch of the next row group (global_prefetch_b8)
        __builtin_prefetch(bias  + rowIdx + (size_t)8 * W_DIM, 0, 0);
        __builtin_prefetch(space + rowIdx + (size_t)8 * W_DIM, 0, 0);
#pragma unroll
        for (int u = 0; u < 8; ++u) {
            const int k = ks + k0 + u;
            const float t  = (sp[u] + b[u] + ((k == l) ? extra : 0.0f)) * xs[k0 + u];
            const float mn = fmaxf(m, t);
            s = s * __expf(m - mn) + __expf(t - mn);
            m = mn;
        }
        rowIdx += (size_t)8 * W_DIM;
    }
    pm[(size_t)blockIdx.y * W_DIM + col] = m;
    ps[(size_t)blockIdx.y * W_DIM + col] = s;
}

// Combine 4 slice stats -> c[col] = m + ln(Z)
__global__ __launch_bounds__(256) void k_combine(
    const float* __restrict__ pm, const float* __restrict__ ps,
    float* __restrict__ cvec, int colBase)
{
    const int col = colBase + blockIdx.x * 256 + threadIdx.x;
    float m0 = pm[col];
    float m1 = pm[W_DIM + col];
    float m2 = pm[2 * W_DIM + col];
    float m3 = pm[3 * W_DIM + col];
    float mm = fmaxf(fmaxf(m0, m1), fmaxf(m2, m3));
    float s  = ps[col]             * __expf(m0 - mm)
             + ps[W_DIM + col]     * __expf(m1 - mm)
             + ps[2 * W_DIM + col] * __expf(m2 - mm)
             + ps[3 * W_DIM + col] * __expf(m3 - mm);
    cvec[col] = mm + __logf(s);
}

// Pass 2: one row k per block, 4096 columns; g = max_col (t - c); wave-reduce; atomicMax.
__global__ __launch_bounds__(256) void k_pass2(
    const float* __restrict__ bias, const float* __restrict__ space,
    const float* __restrict__ kern, const float* __restrict__ x,
    const float* __restrict__ cvec, unsigned* __restrict__ keys, int colBase)
{
    __shared__ float cbuf[P2_TILE];
    const int tid = threadIdx.x;
    const int k   = blockIdx.y;
    const int cb  = colBase + blockIdx.x * P2_TILE;

    // Stage this block's c[] slice (16 KB) into LDS via async global->LDS b128 loads.
#if HAVE_ASYNC_LDS
    {
        gv4i_p gp = (gv4i_p)(cvec + cb + (size_t)tid * 16);
        lv4i_p lp = (lv4i_p)(&cbuf[tid * 16]);
        __builtin_amdgcn_global_load_async_to_lds_b128(gp, lp,  0, 0);
        __builtin_amdgcn_global_load_async_to_lds_b128(gp, lp, 16, 0);
        __builtin_amdgcn_global_load_async_to_lds_b128(gp, lp, 32, 0);
        __builtin_amdgcn_global_load_async_to_lds_b128(gp, lp, 48, 0);
#if __has_builtin(__builtin_amdgcn_s_wait_asynccnt)
        __builtin_amdgcn_s_wait_asynccnt(0);
#else
        asm volatile("s_wait_asynccnt 0" ::: "memory");
#endif
    }
    __syncthreads();
#else
    {
        const float4* src = (const float4*)(cvec + cb);
        float4* dst = (float4*)cbuf;
#pragma unroll
        for (int q = 0; q < 4; ++q) dst[tid + 256 * q] = src[tid + 256 * q];
    }
    __syncthreads();
#endif

    const float xk  = x[k];
    const int col0  = cb + tid * 16;
    const int l0    = col0 % E_DIM;                      // multiple of 16
    const int istar = ((unsigned)(k - l0) < 16u) ? (k - l0) : -1;
    float extra = 0.0f;
    if (istar >= 0) {
        const int colh = col0 + istar;
        extra = kern[(size_t)k * W_DIM + colh] * x[colh / E_DIM] * INV_SQRT2;
    }

    const size_t rowOff = (size_t)k * W_DIM + col0;
    const float4* b4 = (const float4*)(bias  + rowOff);
    const float4* s4 = (const float4*)(space + rowOff);
    const float4* c4 = (const float4*)(&cbuf[tid * 16]);

    float4 bb[4], ss[4];
#pragma unroll
    for (int q = 0; q < 4; ++q) { bb[q] = b4[q]; ss[q] = s4[q]; }

    float g = -3.0e38f;
#pragma unroll
    for (int q = 0; q < 4; ++q) {
        const float4 cc = c4[q];
        float t;
        t = (ss[q].x + bb[q].x + ((q * 4 + 0) == istar ? extra : 0.0f)) * xk; g = fmaxf(g, t - cc.x);
        t = (ss[q].y + bb[q].y + ((q * 4 + 1) == istar ? extra : 0.0f)) * xk; g = fmaxf(g, t - cc.y);
        t = (ss[q].z + bb[q].z + ((q * 4 + 2) == istar ? extra : 0.0f)) * xk; g = fmaxf(g, t - cc.z);
        t = (ss[q].w + bb[q].w + ((q * 4 + 3) == istar ? extra : 0.0f)) * xk; g = fmaxf(g, t - cc.w);
    }

    // wave32 max-reduce, one atomic per wave
#pragma unroll
    for (int off = 16; off > 0; off >>= 1)
        g = fmaxf(g, __shfl_xor(g, off, 32));
    if ((tid & 31) == 0) atomicMax(keys + k, enc_f32(g));
}

__global__ void k_final(float* __restrict__ out) {
    int t = threadIdx.x;
    if (t < E_DIM) {
        unsigned u = ((unsigned*)out)[t];
        out[t] = __expf(dec_f32(u));
    }
}

extern "C" void kernel_launch(void* const* d_in, const int* in_sizes, int n_in,
                              void* d_out, int out_size, void* d_ws, size_t ws_size,
                              hipStream_t stream)
{
    (void)in_sizes; (void)n_in; (void)out_size; (void)ws_size;
    const float* x     = (const float*)d_in[0];
    const float* kern  = (const float*)d_in[1];
    const float* bias  = (const float*)d_in[2];
    const float* space = (const float*)d_in[3];

    unsigned* keys = (unsigned*)d_out;              // d_out doubles as key buffer
    float* ws   = (float*)d_ws;
    float* cvec = ws;                               // W floats      (0.6 MB)
    float* pm   = ws + (size_t)W_DIM;               // 4W floats     (2.25 MB)
    float* ps   = ws + (size_t)5 * W_DIM;           // 4W floats     (2.25 MB)

    k_init<<<1, E_DIM, 0, stream>>>(keys);
    for (int cc = 0; cc < NCHUNK; ++cc) {
        const int base = cc * CHUNK_COLS;
        k_pass1<<<dim3(CHUNK_COLS / 256, NSLICE), 256, 0, stream>>>(
            bias, space, kern, x, pm, ps, base);
        k_combine<<<CHUNK_COLS / 256, 256, 0, stream>>>(pm, ps, cvec, base);
        k_pass2<<<dim3(CHUNK_COLS / P2_TILE, E_DIM), 256, 0, stream>>>(
            bias, space, kern, x, cvec, keys, base);
    }
    k_final<<<1, E_DIM, 0, stream>>>((float*)d_out);
}